// Loss_26233660244742
// MI455X (gfx1250) — compile-verified
//
#include <hip/hip_runtime.h>
#include <stdint.h>

// ---------------- problem constants (from reference) ----------------
#define NC 20            // classes
#define NA 5             // anchors
#define HW 169           // 13*13 cells
#define CH 125           // NA*(5+NC) channels
#define TSTR 25          // target floats per cell
#define TGT_PER_B 4225   // HW*TSTR
#define PRED_PER_B 21125 // CH*HW

typedef int   v4i __attribute__((ext_vector_type(4)));
typedef int   v8i __attribute__((ext_vector_type(8)));
typedef float v2f __attribute__((ext_vector_type(2)));
typedef float v8f __attribute__((ext_vector_type(8)));

// Low 32 bits of a flat LDS-aperture address == LDS byte offset (ISA 10.2).
__device__ __forceinline__ unsigned lds_off(const void* p) {
  return (unsigned)(uintptr_t)p;
}

__device__ __forceinline__ void wait_tensor0() {
#if __has_builtin(__builtin_amdgcn_s_wait_tensorcnt)
  __builtin_amdgcn_s_wait_tensorcnt(0);
#else
  asm volatile("s_wait_tensorcnt 0x0" ::: "memory");
#endif
}

// D# group 0: count=1, lds_addr, 57-bit global addr, type=2 ("image").
__device__ __forceinline__ v4i tdm_g0(unsigned lds, const void* g) {
  unsigned long long ga = (unsigned long long)(uintptr_t)g;
  v4i g0;
  g0[0] = 1;
  g0[1] = (int)lds;
  g0[2] = (int)(unsigned)(ga & 0xFFFFFFFFull);
  g0[3] = (int)((unsigned)((ga >> 32) & 0x1FFFFFFull) | 0x80000000u);
  return g0;
}

// 2D tile of 4-byte elems: tile1 rows of tile0 elems, row stride s0 elems,
// packed contiguously into LDS.  (2-group form: tensors up to 2D.)
__device__ __forceinline__ void tdm_load_2d(void* lds, const void* g,
                                            int tdim0, int tile0,
                                            int tdim1, int tile1, int s0) {
  v4i g0 = tdm_g0(lds_off(lds), g);
  v8i g1;
  g1[0] = 0x00020000;                                             // data_size = 4B
  g1[1] = (int)((tdim0 & 0xFFFF) << 16);                          // tensor_dim0 lo16
  g1[2] = (int)((((unsigned)tdim0 >> 16) & 0xFFFF) | ((tdim1 & 0xFFFF) << 16));
  g1[3] = (int)(((((unsigned)tdim1) >> 16) & 0xFFFF) | ((tile0 & 0xFFFF) << 16));
  g1[4] = (int)(tile1 & 0xFFFF);                                  // tile_dim1 (tile_dim2=0)
  g1[5] = s0;                                                     // tensor_dim0_stride lo32
  g1[6] = 0;
  g1[7] = 0;
  asm volatile("tensor_load_to_lds %0, %1" :: "s"(g0), "s"(g1) : "memory");
}

// 3D tile: x (tile0, stride 1) / y (tile1, stride s0) / z (tile2, stride s1).
__device__ __forceinline__ void tdm_load_3d(void* lds, const void* g,
                                            int tdim0, int tile0,
                                            int tdim1, int tile1, int s0,
                                            int tdim2, int tile2, int s1) {
  v4i g0 = tdm_g0(lds_off(lds), g);
  v8i g1; v4i g2, g3;
  g1[0] = 0x00020000;
  g1[1] = (int)((tdim0 & 0xFFFF) << 16);
  g1[2] = (int)((((unsigned)tdim0 >> 16) & 0xFFFF) | ((tdim1 & 0xFFFF) << 16));
  g1[3] = (int)(((((unsigned)tdim1) >> 16) & 0xFFFF) | ((tile0 & 0xFFFF) << 16));
  g1[4] = (int)((tile1 & 0xFFFF) | ((tile2 & 0xFFFF) << 16));
  g1[5] = s0;                                  // dim0_stride[31:0]
  g1[6] = (int)((s1 & 0xFFFF) << 16);          // dim0_stride[47:32]=0 | dim1_stride[15:0]
  g1[7] = (int)(((unsigned)s1) >> 16);         // dim1_stride[47:16]
  g2[0] = tdim2;                               // tensor_dim2
  g2[1] = 1;                                   // tensor_dim3 (iterate off)
  g2[2] = 0; g2[3] = 0;                        // dim2 stride / tile_dim3 unused
  g3[0] = 0; g3[1] = 0; g3[2] = 0; g3[3] = 0;  // dim3/dim4 unused
  asm volatile("tensor_load_to_lds %0, %1, %2, %3"
               :: "s"(g0), "s"(g1), "s"(g2), "s"(g3) : "memory");
}

__device__ __forceinline__ float sigmoidf(float x) {
  return 1.0f / (1.0f + __expf(-x));
}

// Matches reference _box_iou exactly (areas from (x2-x1)*(y2-y1), +1e-10).
__device__ __forceinline__ float box_iou(float px, float py, float aw, float ah,
                                         float gx, float gy, float gw, float gh) {
  float b1x1 = px - aw * 0.5f, b1x2 = px + aw * 0.5f;
  float b1y1 = py - ah * 0.5f, b1y2 = py + ah * 0.5f;
  float b2x1 = gx - gw * 0.5f, b2x2 = gx + gw * 0.5f;
  float b2y1 = gy - gh * 0.5f, b2y2 = gy + gh * 0.5f;
  float iw = fmaxf(fminf(b1x2, b2x2) - fmaxf(b1x1, b2x1), 0.0f);
  float ih = fmaxf(fminf(b1y2, b2y2) - fmaxf(b1y1, b2y1), 0.0f);
  float inter = iw * ih;
  float a1 = (b1x2 - b1x1) * (b1y2 - b1y1);
  float a2 = (b2x2 - b2x1) * (b2y2 - b2y1);
  return inter / (a1 + a2 - inter + 1e-10f);
}

// Deterministic 256-thread tree reduction.
__device__ __forceinline__ float block_sum(float v, float* red) {
  const int t = threadIdx.x;
  red[t] = v;
  __syncthreads();
#pragma unroll
  for (int s = 128; s > 0; s >>= 1) {
    if (t < s) red[t] += red[t + s];
    __syncthreads();
  }
  return red[0];
}

// ---------------- K0: init first-obj index ----------------
__global__ void init_kernel(int* firstFlat) {
  if (threadIdx.x == 0) *firstFlat = 0x7FFFFFFF;
}

// ---------------- K1: box / conf / noobj loss + first-obj scan ----------------
__global__ __launch_bounds__(256) void yolo_main_kernel(
    const float* __restrict__ pred, const float* __restrict__ tgt,
    const float* __restrict__ anchor, float* __restrict__ partial,
    int* __restrict__ firstFlat) {
  __shared__ float lt[TGT_PER_B];      // target tile  [cell][25]
  __shared__ float lp[NA * 5 * HW];    // pred tile    [anchor][{conf,x,y,w,h}][cell]
  __shared__ float red[256];
  const int b = blockIdx.x;
  const int t = threadIdx.x;

  if (t < 32) {  // wave 0 issues both TDM descriptors (TDM ignores EXEC)
    // channels a*25+20..a*25+24 for a=0..4: x=169 cells, y=5 ch (stride 169),
    // z=5 anchors (stride 25*169)
    tdm_load_3d(lp, pred + (size_t)b * PRED_PER_B + (size_t)NC * HW,
                HW, HW, TSTR, 5, HW, NA, NA, TSTR * HW);
    tdm_load_2d(lt, tgt + (size_t)b * TGT_PER_B, TGT_PER_B, TGT_PER_B, 1, 1,
                TGT_PER_B);
  }
  wait_tensor0();
  __syncthreads();

  float lsum = 0.0f;
  if (t < HW) {
    const float gconf = lt[t * TSTR + 20];
    const float gx = lt[t * TSTR + 21], gy = lt[t * TSTR + 22];
    const float gw = lt[t * TSTR + 23], gh = lt[t * TSTR + 24];
    const bool obj = (gconf != 0.0f);

    float best = -1.0f;
    float bpc = 0.f, bpx = 0.f, bpy = 0.f, bpw = 0.f, bph = 0.f;
    float noobjsum = 0.0f;
#pragma unroll
    for (int a = 0; a < NA; ++a) {
      const float pc = sigmoidf(lp[(a * 5 + 0) * HW + t]);
      const float px = sigmoidf(lp[(a * 5 + 1) * HW + t]);
      const float py = sigmoidf(lp[(a * 5 + 2) * HW + t]);
      const float pw = __expf(lp[(a * 5 + 3) * HW + t]);
      const float ph = __expf(lp[(a * 5 + 4) * HW + t]);
      noobjsum += pc * pc;
      const float iou = box_iou(px, py, pw * anchor[2 * a], ph * anchor[2 * a + 1],
                                gx, gy, gw, gh);
      if (iou > best) {  // strict > == jnp.argmax first-max semantics
        best = iou; bpc = pc; bpx = px; bpy = py; bpw = pw; bph = ph;
      }
    }
    if (obj) {
      noobjsum -= bpc * bpc;                       // masked anchor leaves noobj set
      lsum += (bpc - 1.0f) * (bpc - 1.0f);         // conf loss
      const float dx = bpx - gx, dy = bpy - gy, dw = bpw - gw, dh = bph - gh;
      lsum += 5.0f * (dx * dx + dy * dy + dw * dw + dh * dh);  // LAMBDA_COORD
      atomicMin(firstFlat, b * HW + t);
    }
    lsum += 0.5f * noobjsum;                       // LAMBDA_NOOBJ
  }
  const float tot = block_sum(lsum, red);
  if (t == 0) partial[b] = tot;
}

// ---------------- K2: argmax anchor at first obj cell -> a0 -------------------
// One wave; lanes 0..4 evaluate one anchor each (loads issue in parallel),
// lane 0 resolves the first-max argmax via shuffles.
__global__ __launch_bounds__(32) void pick_anchor_kernel(
    const float* __restrict__ pred, const float* __restrict__ tgt,
    const float* __restrict__ anchor, const int* __restrict__ firstFlat,
    int* __restrict__ a0out) {
  const int lane = threadIdx.x;
  int f = *firstFlat;          // uniform
  if (f == 0x7FFFFFFF) f = 0;  // argmax of all-False == 0
  const int b = f / HW, t = f % HW;
  const float* g = tgt + (size_t)b * TGT_PER_B + (size_t)t * TSTR;
  const float gx = g[21], gy = g[22], gw = g[23], gh = g[24];

  float iou = -2.0f;
  if (lane < NA) {
    const float* p = pred + (size_t)b * PRED_PER_B + t;
    const float px = sigmoidf(p[(lane * TSTR + 21) * HW]);
    const float py = sigmoidf(p[(lane * TSTR + 22) * HW]);
    const float aw = __expf(p[(lane * TSTR + 23) * HW]) * anchor[2 * lane];
    const float ah = __expf(p[(lane * TSTR + 24) * HW]) * anchor[2 * lane + 1];
    iou = box_iou(px, py, aw, ah, gx, gy, gw, gh);
  }
  float best = -1.0f;
  int bi = 0;
#pragma unroll
  for (int a = 0; a < NA; ++a) {       // sequential first-max: jnp.argmax ties
    const float v = __shfl(iou, a, 32);
    if (v > best) { best = v; bi = a; }
  }
  if (lane == 0) *a0out = bi;
}

// ---------------- K3: class cross-entropy (anchor a0) ----------------
__global__ __launch_bounds__(256) void cls_kernel(
    const float* __restrict__ pred, const float* __restrict__ tgt,
    const int* __restrict__ a0p, float* __restrict__ partial) {
  __shared__ float lt[TGT_PER_B];  // target tile
  __shared__ float lg[NC * HW];    // logits for anchor a0: [class][cell]
  __shared__ float red[256];
  const int b = blockIdx.x;
  const int t = threadIdx.x;
  const int a0 = *a0p;  // uniform

  if (t < 32) {
    tdm_load_2d(lg, pred + (size_t)b * PRED_PER_B + (size_t)(a0 * TSTR) * HW,
                HW, HW, NC, NC, HW);
    tdm_load_2d(lt, tgt + (size_t)b * TGT_PER_B, TGT_PER_B, TGT_PER_B, 1, 1,
                TGT_PER_B);
  }
  wait_tensor0();
  __syncthreads();

  float lsum = 0.0f;
  if (t < HW && lt[t * TSTR + 20] != 0.0f) {
    // label = argmax of one-hot target classes
    int label = 0;
    float bv = lt[t * TSTR + 0];
#pragma unroll
    for (int j = 1; j < NC; ++j) {
      const float v = lt[t * TSTR + j];
      if (v > bv) { bv = v; label = j; }
    }
    // log-softmax CE: lse - logit[label]
    float m = lg[t];
#pragma unroll
    for (int j = 1; j < NC; ++j) m = fmaxf(m, lg[j * HW + t]);
    float s = 0.0f;
#pragma unroll
    for (int j = 0; j < NC; ++j) s += __expf(lg[j * HW + t] - m);
    lsum = m + __logf(s) - lg[label * HW + t];
  }
  const float tot = block_sum(lsum, red);
  if (t == 0) partial[b] = tot;
}

// ---------------- K4: matrix-core final reduction ----------------
// Ones-matrix WMMA reduction: A = ones(16x4)  =>  D[m][n] = sum_k B[k][n] + C[m][n].
// Each iteration folds 64 partials (any lane<->element bijection is valid since
// A is uniform) into 16 running f32 column sums; every row of D is identical.
// One wave, EXEC all-ones (WMMA requirement), fixed order => deterministic.
__global__ __launch_bounds__(32) void reduce_kernel(const float* __restrict__ p1,
                                                    const float* __restrict__ p2,
                                                    int nb, float* __restrict__ out) {
  const int lane = threadIdx.x;
  const int n2 = 2 * nb;
#if __has_builtin(__builtin_amdgcn_wmma_f32_16x16x4_f32)
  v8f acc = {};
  const v2f ones = {1.0f, 1.0f};
  for (int base = 0; base < n2; base += 64) {
    const int i0 = base + lane;
    const int i1 = base + 32 + lane;
    v2f bmat;
    bmat[0] = (i0 < n2) ? ((i0 < nb) ? p1[i0] : p2[i0 - nb]) : 0.0f;
    bmat[1] = (i1 < n2) ? ((i1 < nb) ? p1[i1] : p2[i1 - nb]) : 0.0f;
    acc = __builtin_amdgcn_wmma_f32_16x16x4_f32(
        false, ones, false, bmat, (short)0, acc, false, false);
  }
  // Row 0 of D lives in acc[0]: lanes 0-15 = N0..15, lanes 16-31 repeat row 8
  // (identical).  Butterfly-sum the 16-lane half.
  float v = acc[0];
#pragma unroll
  for (int off = 8; off > 0; off >>= 1) v += __shfl_xor(v, off, 32);
  if (lane == 0) out[0] = v;
#else
  // Scalar fallback: lane-strided deterministic sums + shuffle tree.
  float s = 0.0f;
  for (int i = lane; i < nb; i += 32) s += p1[i];
  for (int i = lane; i < nb; i += 32) s += p2[i];
#pragma unroll
  for (int off = 16; off > 0; off >>= 1) s += __shfl_xor(s, off, 32);
  if (lane == 0) out[0] = s;
#endif
}

extern "C" void kernel_launch(void* const* d_in, const int* in_sizes, int n_in,
                              void* d_out, int out_size, void* d_ws, size_t ws_size,
                              hipStream_t stream) {
  const float* pred = (const float*)d_in[0];
  const float* tgt = (const float*)d_in[1];
  const float* anchor = (const float*)d_in[2];
  const int nb = in_sizes[1] / TGT_PER_B;  // batch (2048)

  char* ws = (char*)d_ws;
  int* firstFlat = (int*)ws;                       // 4 B
  int* a0 = (int*)(ws + 16);                       // 4 B
  float* p1 = (float*)(ws + 64);                   // nb floats
  float* p2 = (float*)(ws + 64 + (size_t)nb * 4);  // nb floats

  init_kernel<<<1, 32, 0, stream>>>(firstFlat);
  yolo_main_kernel<<<nb, 256, 0, stream>>>(pred, tgt, anchor, p1, firstFlat);
  pick_anchor_kernel<<<1, 32, 0, stream>>>(pred, tgt, anchor, firstFlat, a0);
  cls_kernel<<<nb, 256, 0, stream>>>(pred, tgt, a0, p2);
  reduce_kernel<<<1, 32, 0, stream>>>(p1, p2, nb, (float*)d_out);
}